// GCN_21328807592611
// MI455X (gfx1250) — compile-verified
//
#include <hip/hip_runtime.h>

// ---------------------------------------------------------------------------
// GCN (4x DGL GraphConv, norm='both') for MI455X / gfx1250.
// SpMM is the bandwidth hog (L2-resident atomics); GEMM uses f32 WMMA
// (V_WMMA_F32_16X16X4_F32) with an LDS-staged A tile to stay
// reference-precision while keeping the matrix pipe fed.
// ---------------------------------------------------------------------------

typedef __attribute__((ext_vector_type(2))) float v2f;
typedef __attribute__((ext_vector_type(8))) float v8f;

#define HDIM 128
#define LDS_STRIDE 132   // 128 + 4 pad: lanes q=0..15 hit banks 4q.. -> conflict-free

// ---------------- zero-fill (float4 grid-stride) ----------------
__global__ void gcn_zero_kernel(float4* __restrict__ p, long n4) {
    long i = (long)blockIdx.x * blockDim.x + threadIdx.x;
    long stride = (long)gridDim.x * blockDim.x;
    float4 z; z.x = 0.f; z.y = 0.f; z.z = 0.f; z.w = 0.f;
    for (; i < n4; i += stride) p[i] = z;
}

// ---------------- degree accumulation ----------------
__global__ void gcn_degree_kernel(const int* __restrict__ src,
                                  const int* __restrict__ dst,
                                  float* __restrict__ outdeg,
                                  float* __restrict__ indeg, int E) {
    int e = blockIdx.x * blockDim.x + threadIdx.x;
    if (e < E) {
        atomicAdd(&outdeg[src[e]], 1.0f);
        atomicAdd(&indeg[dst[e]], 1.0f);
    }
}

// deg -> rsqrt(max(deg,1)) in place
__global__ void gcn_inv_kernel(float* __restrict__ a, float* __restrict__ b, int n) {
    int i = blockIdx.x * blockDim.x + threadIdx.x;
    if (i < n) {
        a[i] = rsqrtf(fmaxf(a[i], 1.0f));
        b[i] = rsqrtf(fmaxf(b[i], 1.0f));
    }
}

// ---------------- pad last-layer weights: 128x10 -> 128x16, bias 10 -> 16 ----
__global__ void gcn_pad_w_kernel(const float* __restrict__ W3,
                                 const float* __restrict__ b3,
                                 float* __restrict__ Wp, float* __restrict__ bp,
                                 int ocols) {
    int t = blockIdx.x * blockDim.x + threadIdx.x;
    if (t < HDIM * 16) {
        int k = t >> 4, n = t & 15;
        Wp[t] = (n < ocols) ? W3[k * ocols + n] : 0.0f;
    }
    if (t < 16) bp[t] = (t < ocols) ? b3[t] : 0.0f;
}

// ---------------- SpMM: one wave (32 lanes x float4) per edge ----------------
// agg[dst[e]] += X[src[e]] * out_inv[src[e]]
__global__ void __launch_bounds__(256)
gcn_spmm_kernel(const float* __restrict__ X, const int* __restrict__ src,
                const int* __restrict__ dst, const float* __restrict__ out_inv,
                float* __restrict__ agg, int E) {
    long t = (long)blockIdx.x * blockDim.x + threadIdx.x;
    int e = (int)(t >> 5);
    if (e >= E) return;
    int c = (int)(t & 31) << 2;                 // 0,4,...,124
    int s = src[e];
    int d = dst[e];
    float sc = out_inv[s];
    float4 v = *(const float4*)(X + (long)s * HDIM + c);
    float* o = agg + (long)d * HDIM + c;
    atomicAdd(o + 0, v.x * sc);
    atomicAdd(o + 1, v.y * sc);
    atomicAdd(o + 2, v.z * sc);
    atomicAdd(o + 3, v.w * sc);
}

// ---------------- GEMM: (agg * in_inv) @ W + b, optional ReLU ----------------
// grid.x = N/16 row tiles; block = 256 threads = 8 waves.
// A tile (16 x 128, scaled by in_inv) staged in LDS once per block; wave w
// computes 16x16 output tile at columns [16w, 16w+16). NC = B row stride
// (compile-time): 128 for hidden layers, 16 for the padded last layer.
template <int NC>
__global__ void __launch_bounds__(256)
gcn_gemm_wmma_kernel(const float* __restrict__ A, const float* __restrict__ in_inv,
                     const float* __restrict__ W, const float* __restrict__ bias,
                     float* __restrict__ Out, int outcols, int nColTiles, int doRelu) {
    __shared__ float As[16 * LDS_STRIDE];

    const int rowBase = blockIdx.x << 4;

    // cooperative load + scale: 2048 floats / 256 threads = 8 contiguous each
    {
        const int t = threadIdx.x;
        const int r = t >> 4;               // row 0..15 (16 threads per row)
        const int cb = (t & 15) << 3;       // col 0,8,...,120
        const float s = in_inv[rowBase + r];
        const float4* g = (const float4*)(A + (long)(rowBase + r) * HDIM + cb);
        float4 v0 = g[0], v1 = g[1];
        v0.x *= s; v0.y *= s; v0.z *= s; v0.w *= s;
        v1.x *= s; v1.y *= s; v1.z *= s; v1.w *= s;
        float4* sh = (float4*)(&As[r * LDS_STRIDE + cb]);
        sh[0] = v0; sh[1] = v1;
    }
    __syncthreads();

    const int lane = threadIdx.x & 31;
    const int wave = threadIdx.x >> 5;
    if (wave >= nColTiles) return;

    const int colBase = wave << 4;
    const int q    = lane & 15;
    const int half = lane >> 4;
    const int n = colBase + q;              // B/D column held by this lane

    v8f c;
    const float cinit = bias[n];
#pragma unroll
    for (int r = 0; r < 8; ++r) c[r] = cinit;

    const float* arow = As + q * LDS_STRIDE + (half << 1);   // LDS, imm offsets
    const float* wcol = W + (half << 1) * NC + n;            // global, imm offsets

#pragma unroll
    for (int kk = 0; kk < 32; ++kk) {
        v2f a, b;
        float2 av = *(const float2*)(arow + (kk << 2));      // ds_load_b64
        a.x = av.x; a.y = av.y;
        b.x = wcol[(kk << 2) * NC];
        b.y = wcol[(kk << 2) * NC + NC];
        c = __builtin_amdgcn_wmma_f32_16x16x4_f32(false, a, false, b,
                                                  (short)0, c, false, false);
    }

    if (n < outcols) {
#pragma unroll
        for (int r = 0; r < 8; ++r) {
            float v = c[r];
            if (doRelu) v = fmaxf(v, 0.0f);
            Out[(long)(rowBase + r + (half << 3)) * outcols + n] = v;
        }
    }
}

// ---------------------------------------------------------------------------
extern "C" void kernel_launch(void* const* d_in, const int* in_sizes, int n_in,
                              void* d_out, int out_size, void* d_ws, size_t ws_size,
                              hipStream_t stream) {
    const float* feats = (const float*)d_in[0];
    const int*   src   = (const int*)d_in[1];
    const int*   dst   = (const int*)d_in[2];
    const float* Wm[4] = {(const float*)d_in[3], (const float*)d_in[5],
                          (const float*)d_in[7], (const float*)d_in[9]};
    const float* Bm[4] = {(const float*)d_in[4], (const float*)d_in[6],
                          (const float*)d_in[8], (const float*)d_in[10]};

    const int E = in_sizes[1];
    const int N = in_sizes[0] / HDIM;
    const int O = out_size / N;                          // 10

    // workspace layout
    float* out_inv = (float*)d_ws;                       // N
    float* in_inv  = out_inv + N;                        // N
    float* X       = in_inv + N;                         // N*128
    float* AGG     = X + (size_t)N * HDIM;               // N*128
    float* Wp      = AGG + (size_t)N * HDIM;             // 128*16 (padded W3)
    float* bp      = Wp + HDIM * 16;                     // 16    (padded b3)

    // degrees -> inverse sqrt norms; pad last-layer weights
    {
        long n4 = (long)(2 * N) / 4;
        gcn_zero_kernel<<<512, 256, 0, stream>>>((float4*)out_inv, n4);
        gcn_degree_kernel<<<(E + 255) / 256, 256, 0, stream>>>(src, dst, out_inv, in_inv, E);
        gcn_inv_kernel<<<(N + 255) / 256, 256, 0, stream>>>(out_inv, in_inv, N);
        gcn_pad_w_kernel<<<(HDIM * 16 + 255) / 256, 256, 0, stream>>>(Wm[3], Bm[3], Wp, bp, O);
    }

    const int rowTiles = N / 16;                 // 3125 for N=50000
    const long aggN4   = (long)N * HDIM / 4;
    const long spmmBlocks = ((long)E * 32 + 255) / 256;

    for (int l = 0; l < 4; ++l) {
        gcn_zero_kernel<<<4096, 256, 0, stream>>>((float4*)AGG, aggN4);

        const float* Xin = (l == 0) ? feats : X;
        gcn_spmm_kernel<<<(int)spmmBlocks, 256, 0, stream>>>(Xin, src, dst, out_inv, AGG, E);

        if (l < 3) {
            gcn_gemm_wmma_kernel<HDIM><<<rowTiles, 256, 0, stream>>>(
                AGG, in_inv, Wm[l], Bm[l], X, HDIM, 8, 1);
        } else {
            gcn_gemm_wmma_kernel<16><<<rowTiles, 256, 0, stream>>>(
                AGG, in_inv, Wp, bp, (float*)d_out, O, 1, 0);
        }
    }
}